// MSELoss_28741921145432
// MI455X (gfx1250) — compile-verified
//
#include <hip/hip_runtime.h>

// Vector types for wide loads and WMMA operands.
typedef __attribute__((ext_vector_type(4))) float f4;
typedef __attribute__((ext_vector_type(2))) float v2f;  // A/B of V_WMMA_F32_16X16X4_F32 (2 VGPRs)
typedef __attribute__((ext_vector_type(8))) float v8f;  // C/D (8 VGPRs)

// Full-wave (32-lane) f32 sum using V_WMMA_F32_16X16X4_F32 with B = ones.
// A layout (16x4 f32): lanes 0-15 hold {K=0 (vgpr0), K=1 (vgpr1)} for M=lane,
// lanes 16-31 hold {K=2, K=3}. We place acc in vgpr0 and 0 in vgpr1, so
// rowsum(m) = acc(lane m) + acc(lane m+16). D = A x ones: every column equals
// the row sums; lanes 0-15 carry rows 0-7 in d[0..7], lanes 16-31 rows 8-15.
// Summing d[0..7] then xor-16 add yields the total in every lane.
// EXEC must be all-1s at this call site (no divergence around it).
__device__ __forceinline__ float wave_sum_f32_wmma(float acc) {
    v2f a; a[0] = acc;  a[1] = 0.0f;
    v2f b; b[0] = 1.0f; b[1] = 1.0f;
    v8f c = {};
    v8f d = __builtin_amdgcn_wmma_f32_16x16x4_f32(
        /*neg_a=*/false, a, /*neg_b=*/false, b,
        /*c_mod=*/(short)0, c, /*reuse_a=*/false, /*reuse_b=*/false);
    float s = ((d[0] + d[1]) + (d[2] + d[3])) + ((d[4] + d[5]) + (d[6] + d[7]));
    s += __shfl_xor(s, 16, 32);
    return s;
}

__global__ void mse_partial_kernel(const float* __restrict__ d1,
                                   const float* __restrict__ d2,
                                   float* __restrict__ partial,
                                   long long n) {
    const long long n4     = n >> 2;
    const long long stride = (long long)gridDim.x * blockDim.x;
    const long long tid    = (long long)blockIdx.x * blockDim.x + threadIdx.x;

    const f4* __restrict__ a4 = (const f4*)d1;
    const f4* __restrict__ b4 = (const f4*)d2;

    float acc = 0.0f;
    // Streaming main loop: 2 x 16B nontemporal loads per iteration.
    for (long long i = tid; i < n4; i += stride) {
        f4 a = __builtin_nontemporal_load(&a4[i]);
        f4 b = __builtin_nontemporal_load(&b4[i]);
        float d0 = a[0] - b[0];
        float d1v = a[1] - b[1];
        float d2v = a[2] - b[2];
        float d3 = a[3] - b[3];
        acc = fmaf(d0, d0, acc);
        acc = fmaf(d1v, d1v, acc);
        acc = fmaf(d2v, d2v, acc);
        acc = fmaf(d3, d3, acc);
    }
    // Scalar tail (n not multiple of 4).
    for (long long i = (n4 << 2) + tid; i < n; i += stride) {
        float dv = d1[i] - d2[i];
        acc = fmaf(dv, dv, acc);
    }

    // Wave reduction via WMMA (uniform control flow here; EXEC all-1s).
    float wsum = wave_sum_f32_wmma(acc);

    __shared__ float lds[32];
    const int lane = threadIdx.x & 31;
    const int wave = threadIdx.x >> 5;
    if (lane == 0) lds[wave] = wsum;
    __syncthreads();
    if (threadIdx.x == 0) {
        const int nwaves = blockDim.x >> 5;
        float s = 0.0f;
        for (int w = 0; w < nwaves; ++w) s += lds[w];
        partial[blockIdx.x] = s;
    }
}

__global__ void mse_final_kernel(const float* __restrict__ partial,
                                 int nblocks,
                                 float* __restrict__ out) {
    float acc = 0.0f;
    for (int i = threadIdx.x; i < nblocks; i += blockDim.x) acc += partial[i];

    float wsum = wave_sum_f32_wmma(acc);

    __shared__ float lds[32];
    const int lane = threadIdx.x & 31;
    const int wave = threadIdx.x >> 5;
    if (lane == 0) lds[wave] = wsum;
    __syncthreads();
    if (threadIdx.x == 0) {
        const int nwaves = blockDim.x >> 5;
        float s = 0.0f;
        for (int w = 0; w < nwaves; ++w) s += lds[w];
        out[0] = 0.5f * s;  // loss = 0.5 * sum((d1-d2)^2)
    }
}

extern "C" void kernel_launch(void* const* d_in, const int* in_sizes, int n_in,
                              void* d_out, int out_size, void* d_ws, size_t ws_size,
                              hipStream_t stream) {
    const float* d1 = (const float*)d_in[0];
    const float* d2 = (const float*)d_in[1];
    float* out      = (float*)d_out;
    float* partial  = (float*)d_ws;

    const long long n = (long long)in_sizes[0];  // 40,000,000 f32 per input

    const int threads = 256;  // 8 wave32s per block
    int blocks = 4096;        // ~1M lanes in flight: enough to saturate 23.3 TB/s

    // Keep per-block partials inside the workspace.
    long long ws_cap = (long long)(ws_size / sizeof(float));
    if (ws_cap < 1) ws_cap = 1;
    if (blocks > ws_cap) blocks = (int)ws_cap;

    // Don't launch more blocks than there is vectorized work.
    long long work4 = (n + 3) >> 2;
    long long max_blocks = (work4 + threads - 1) / threads;
    if (max_blocks < 1) max_blocks = 1;
    if (blocks > max_blocks) blocks = (int)max_blocks;
    if (blocks < 1) blocks = 1;

    mse_partial_kernel<<<blocks, threads, 0, stream>>>(d1, d2, partial, n);
    mse_final_kernel<<<1, 256, 0, stream>>>(partial, blocks, out);
}